// TripletSemihardLoss_3813930959505
// MI455X (gfx1250) — compile-verified
//
#include <hip/hip_runtime.h>
#include <math.h>

typedef __attribute__((ext_vector_type(2))) float v2f;
typedef __attribute__((ext_vector_type(8))) float v8f;

#define BN 384   // batch
#define DIM 512  // embedding dim
#define NT 24    // 16x16 tiles per matrix dim (384/16)

// ---------------------------------------------------------------------------
// Kernel A: sq[i] = ||V[i]||^2.  One wave per row, lane-strided, shfl-reduce.
// ---------------------------------------------------------------------------
__global__ void sq_kernel(const float* __restrict__ V, float* __restrict__ sq) {
    int wave = (blockIdx.x * blockDim.x + threadIdx.x) >> 5;
    int lane = threadIdx.x & 31;
    if (wave >= BN) return;
    const float* row = V + (size_t)wave * DIM;
    float s = 0.0f;
    for (int c = lane; c < DIM; c += 32) {
        float x = row[c];
        s = fmaf(x, x, s);
    }
    #pragma unroll
    for (int off = 16; off >= 1; off >>= 1) s += __shfl_xor(s, off, 32);
    if (lane == 0) sq[wave] = s;
}

// ---------------------------------------------------------------------------
// Kernel B: Gram via V_WMMA_F32_16X16X4_F32, then convert tile to Euclidean
// distance (matching the reference's relu/err/sqrt/zero-diagonal semantics).
// One wave owns one 16x16 output tile; 8 waves per block; 576 tiles total.
//
// Fragment layout (ISA 7.12.2, 32-bit A 16x4):
//   lane<16 : row = base+lane,    holds K = {k0, k0+1}
//   lane>=16: row = base+lane-16, holds K = {k0+2, k0+3}
// B (4x16, lane = N column) mirrors this, so for Gram (B = A-of-other-rows
// transposed) both fragments use the same load pattern.
// ---------------------------------------------------------------------------
__global__ void gram_dist_kernel(const float* __restrict__ V,
                                 const float* __restrict__ sq,
                                 float* __restrict__ dmat) {
    int lane = threadIdx.x & 31;
    int wid  = threadIdx.x >> 5;
    int tile = blockIdx.x * (blockDim.x >> 5) + wid;
    if (tile >= NT * NT) return;
    int ti = tile / NT, tj = tile % NT;
    int i0 = ti * 16, j0 = tj * 16;

    int hv = lane >> 4;       // lane half: 0 or 1
    int r  = lane & 15;       // row within tile
    const float* arow = V + (size_t)(i0 + r) * DIM;
    const float* brow = V + (size_t)(j0 + r) * DIM;

    v8f acc = {};
    for (int k0 = 0; k0 < DIM; k0 += 4) {
        int kk = k0 + 2 * hv;
        v2f a = *(const v2f*)(arow + kk);   // global_load_b64
        v2f b = *(const v2f*)(brow + kk);
        acc = __builtin_amdgcn_wmma_f32_16x16x4_f32(
            /*neg_a=*/false, a, /*neg_b=*/false, b,
            /*c_mod=*/(short)0, acc, /*reuse_a=*/false, /*reuse_b=*/false);
    }

    int j = j0 + r;
    float sqj = sq[j];
    #pragma unroll
    for (int v = 0; v < 8; ++v) {
        int i = i0 + v + 8 * hv;
        float g  = acc[v];
        float d2 = sq[i] + sqj - 2.0f * g;
        float dist = (d2 <= 0.0f) ? 0.0f : sqrtf(d2);
        if (i == j) dist = 0.0f;
        dmat[(size_t)i * BN + j] = dist;
    }
}

// ---------------------------------------------------------------------------
// Kernel C: semihard mining per row i (one block per row, thread j per column).
// negd[k] = label[k]!=label[i] ? d[i,k] : -1  (d >= 0, so -1 never passes the
// "> dij" test and never wins the max).  Each thread scans LDS with float4.
// ---------------------------------------------------------------------------
__global__ void semihard_kernel(const float* __restrict__ dmat,
                                const int* __restrict__ labels,
                                float* __restrict__ rowsum,
                                float* __restrict__ rowcnt) {
    __shared__ float negd[BN];
    __shared__ float psum[12];
    __shared__ float pcnt[12];

    int i = blockIdx.x;
    int j = threadIdx.x;
    int li = labels[i];
    int lj = labels[j];
    float dij = dmat[(size_t)i * BN + j];
    negd[j] = (lj != li) ? dij : -1.0f;
    __syncthreads();

    float minv = 3.4028235e38f;
    float maxv = -1.0f;
    const float4* n4 = (const float4*)negd;
    #pragma unroll 4
    for (int k4 = 0; k4 < BN / 4; ++k4) {
        float4 v = n4[k4];
        maxv = fmaxf(maxv, fmaxf(fmaxf(v.x, v.y), fmaxf(v.z, v.w)));
        if (v.x > dij) minv = fminf(minv, v.x);
        if (v.y > dij) minv = fminf(minv, v.y);
        if (v.z > dij) minv = fminf(minv, v.z);
        if (v.w > dij) minv = fminf(minv, v.w);
    }
    float neg_in = fmaxf(maxv, 0.0f);                 // row_min == 0 in reference
    bool  found  = (minv < 3.0e38f);
    float semihard = found ? minv : neg_in;

    bool  pos     = (lj == li) && (j != i);
    float contrib = pos ? fmaxf(dij - semihard + 1.0f, 0.0f) : 0.0f;
    float cnt     = pos ? 1.0f : 0.0f;

    #pragma unroll
    for (int off = 16; off >= 1; off >>= 1) {
        contrib += __shfl_xor(contrib, off, 32);
        cnt     += __shfl_xor(cnt,     off, 32);
    }
    int lane = j & 31, wid = j >> 5;
    if (lane == 0) { psum[wid] = contrib; pcnt[wid] = cnt; }
    __syncthreads();
    if (j == 0) {
        float s = 0.0f, c = 0.0f;
        #pragma unroll
        for (int w = 0; w < 12; ++w) { s += psum[w]; c += pcnt[w]; }
        rowsum[i] = s;
        rowcnt[i] = c;
    }
}

// ---------------------------------------------------------------------------
// Kernel D: deterministic final reduction over 384 row partials.
// ---------------------------------------------------------------------------
__global__ void finalize_kernel(const float* __restrict__ rowsum,
                                const float* __restrict__ rowcnt,
                                float* __restrict__ out) {
    __shared__ float psum[12];
    __shared__ float pcnt[12];
    int t = threadIdx.x;
    float s = rowsum[t];
    float c = rowcnt[t];
    #pragma unroll
    for (int off = 16; off >= 1; off >>= 1) {
        s += __shfl_xor(s, off, 32);
        c += __shfl_xor(c, off, 32);
    }
    int lane = t & 31, wid = t >> 5;
    if (lane == 0) { psum[wid] = s; pcnt[wid] = c; }
    __syncthreads();
    if (t == 0) {
        float ts = 0.0f, tc = 0.0f;
        #pragma unroll
        for (int w = 0; w < 12; ++w) { ts += psum[w]; tc += pcnt[w]; }
        out[0] = ts / tc;
    }
}

// ---------------------------------------------------------------------------
extern "C" void kernel_launch(void* const* d_in, const int* in_sizes, int n_in,
                              void* d_out, int out_size, void* d_ws, size_t ws_size,
                              hipStream_t stream) {
    (void)in_sizes; (void)n_in; (void)out_size; (void)ws_size;
    const float* V      = (const float*)d_in[0];   // [384,512] f32
    const int*   labels = (const int*)d_in[1];     // [384] i32

    float* ws     = (float*)d_ws;
    float* sq     = ws;                       // 384
    float* dmat   = sq + BN;                  // 384*384
    float* rowsum = dmat + (size_t)BN * BN;   // 384
    float* rowcnt = rowsum + BN;              // 384

    sq_kernel       <<<96, 128, 0, stream>>>(V, sq);
    gram_dist_kernel<<<NT * NT / 8, 256, 0, stream>>>(V, sq, dmat);
    semihard_kernel <<<BN, BN, 0, stream>>>(dmat, labels, rowsum, rowcnt);
    finalize_kernel <<<1, BN, 0, stream>>>(rowsum, rowcnt, (float*)d_out);
}